// GFNN_24550033064031
// MI455X (gfx1250) — compile-verified
//
#include <hip/hip_runtime.h>
#include <hip/hip_bf16.h>

typedef __attribute__((ext_vector_type(2))) float v2f;
typedef __attribute__((ext_vector_type(8))) float v8f;

// ---------------------------------------------------------------------------
// WMMA f32 GEMM with bias (+ optional ReLU on A):  C[M,N] = act(A)[M,K] * B[K,N] + bias
// One wave computes a 16x64 tile of C using V_WMMA_F32_16X16X4_F32.
// Fragment layouts per CDNA5 ISA 7.12.2 (32-bit A 16x4, B 4x16, C/D 16x16):
//   A: lanes 0-15 hold row M=lane, v0=K+0,v1=K+1 ; lanes 16-31 v0=K+2,v1=K+3
//   B: lanes 0-15 hold col N=lane, v0=K+0,v1=K+1 ; lanes 16-31 v0=K+2,v1=K+3
//   C/D: vgpr r, lane l -> M = r + 8*(l>>4), N = l&15
// Guards are wave-uniform => EXEC is all ones at every WMMA.
// Requires M % 16 == 0 (true here: 100000 = 6250*16).
// ---------------------------------------------------------------------------
template <int KDIM, int NDIM, bool RELU>
__global__ __launch_bounds__(256) void wmma_gemm_bias(
    const float* __restrict__ A, const float* __restrict__ B,
    const float* __restrict__ bias, float* __restrict__ C, int M) {
  const int lane   = threadIdx.x & 31;
  const int waveId = threadIdx.x >> 5;
  const int lane16 = lane & 15;
  const int half   = lane >> 4;              // 0 or 1

  constexpr int colTiles = NDIM / 64;
  const int rowTiles = M / 16;
  const int wg = blockIdx.x * 8 + waveId;    // global wave index
  if (wg >= rowTiles * colTiles) return;     // wave-uniform exit

  const int m0 = (wg / colTiles) * 16;
  const int n0 = (wg % colTiles) * 64;

  v8f acc[4] = {};                           // 4 x 16x16 f32 accumulators

  const float* __restrict__ arow = A + (size_t)(m0 + lane16) * KDIM;

  for (int k = 0; k < KDIM; k += 4) {
    const int kb = k + (half << 1);
    float ax = arow[kb];
    float ay = arow[kb + 1];
    if (RELU) { ax = fmaxf(ax, 0.0f); ay = fmaxf(ay, 0.0f); }
    v2f a; a.x = ax; a.y = ay;

    const float* __restrict__ b0p = B + (size_t)kb * NDIM + n0 + lane16;
    const float* __restrict__ b1p = b0p + NDIM;
#pragma unroll
    for (int j = 0; j < 4; ++j) {
      v2f b; b.x = b0p[j * 16]; b.y = b1p[j * 16];
      acc[j] = __builtin_amdgcn_wmma_f32_16x16x4_f32(
          /*neg_a=*/false, a, /*neg_b=*/false, b,
          /*c_mod=*/(short)0, acc[j], /*reuse_a=*/false, /*reuse_b=*/false);
    }
  }

#pragma unroll
  for (int j = 0; j < 4; ++j) {
    const int n = n0 + j * 16 + lane16;
    const float bv = bias[n];
#pragma unroll
    for (int r = 0; r < 8; ++r) {
      const int m = m0 + r + (half << 3);
      C[(size_t)m * NDIM + n] = acc[j][r] + bv;
    }
  }
}

// ---------------------------------------------------------------------------
// SPMM scatter:  dst[row[e], :] += val[e] * src[col[e], :]   (D = 256)
// One thread per (edge, 4-float chunk): coalesced float4 gather of the source
// row; hardware f32 atomics (global_atomic_add_f32) for the scatter.
// ---------------------------------------------------------------------------
__global__ __launch_bounds__(256) void spmm_atomic(
    const int* __restrict__ erow, const int* __restrict__ ecol,
    const float* __restrict__ eval, const float* __restrict__ src,
    float* __restrict__ dst, long long nEdges) {
  const long long gid = (long long)blockIdx.x * blockDim.x + threadIdx.x;
  if (gid >= nEdges * 64) return;
  const int e = (int)(gid >> 6);             // 64 chunks of 4 floats = 256 dims
  const int c = (int)(gid & 63);
  const int r  = erow[e];
  const int cl = ecol[e];
  const float v = eval[e];

  const float4 h = *((const float4*)(src + (size_t)cl * 256) + c);
  float* __restrict__ o = dst + (size_t)r * 256 + (size_t)c * 4;
  __hip_atomic_fetch_add(o + 0, v * h.x, __ATOMIC_RELAXED, __HIP_MEMORY_SCOPE_AGENT);
  __hip_atomic_fetch_add(o + 1, v * h.y, __ATOMIC_RELAXED, __HIP_MEMORY_SCOPE_AGENT);
  __hip_atomic_fetch_add(o + 2, v * h.z, __ATOMIC_RELAXED, __HIP_MEMORY_SCOPE_AGENT);
  __hip_atomic_fetch_add(o + 3, v * h.w, __ATOMIC_RELAXED, __HIP_MEMORY_SCOPE_AGENT);
}

// ---------------------------------------------------------------------------
// Zero-fill (float4 granularity) -- accumulators must be zeroed every call.
// ---------------------------------------------------------------------------
__global__ __launch_bounds__(256) void zero_f4(float4* __restrict__ p, long long n4) {
  const long long i = (long long)blockIdx.x * blockDim.x + threadIdx.x;
  if (i < n4) p[i] = make_float4(0.f, 0.f, 0.f, 0.f);
}

extern "C" void kernel_launch(void* const* d_in, const int* in_sizes, int n_in,
                              void* d_out, int out_size, void* d_ws, size_t ws_size,
                              hipStream_t stream) {
  const float* x    = (const float*)d_in[0];  // [N, 512]
  const int*   erow = (const int*)  d_in[1];  // [E]
  const int*   ecol = (const int*)  d_in[2];  // [E]
  const float* eval = (const float*)d_in[3];  // [E]
  const float* W0   = (const float*)d_in[4];  // [512, 256]
  const float* b0   = (const float*)d_in[5];  // [256]
  const float* W1   = (const float*)d_in[6];  // [256, 128]
  const float* b1   = (const float*)d_in[7];  // [128]
  float* out = (float*)d_out;                 // [N, 128]

  const int N = in_sizes[0] / 512;            // 100000 (divisible by 16)
  const long long E = in_sizes[1];            // 3.2M

  float* h0 = (float*)d_ws;                   // [N, 256]
  float* h1 = h0 + (size_t)N * 256;           // [N, 256]

  const long long n4 = (long long)N * 256 / 4;
  const int zeroBlocks = (int)((n4 + 255) / 256);
  const long long spmmThreads = E * 64;
  const int spmmBlocks = (int)((spmmThreads + 255) / 256);

  // 1) h0 = x @ W0 + b0
  {
    const int waves = (N / 16) * (256 / 64);
    wmma_gemm_bias<512, 256, false><<<(waves + 7) / 8, 256, 0, stream>>>(x, W0, b0, h0, N);
  }
  // 2) h1 = adj @ h0
  zero_f4<<<zeroBlocks, 256, 0, stream>>>((float4*)h1, n4);
  spmm_atomic<<<spmmBlocks, 256, 0, stream>>>(erow, ecol, eval, h0, h1, E);
  // 3) h0 = adj @ h1
  zero_f4<<<zeroBlocks, 256, 0, stream>>>((float4*)h0, n4);
  spmm_atomic<<<spmmBlocks, 256, 0, stream>>>(erow, ecol, eval, h1, h0, E);
  // 4) out = relu(h0) @ W1 + b1
  {
    const int waves = (N / 16) * (128 / 64);
    wmma_gemm_bias<256, 128, true><<<(waves + 7) / 8, 256, 0, stream>>>(h0, W1, b1, out, N);
  }
}